// SmoothingLayer_64407329571483
// MI455X (gfx1250) — compile-verified
//
#include <hip/hip_runtime.h>
#include <cstdint>
#include <cmath>

// SmoothingLayer: 5x5 reflect-padded median filter -> 5x5 zero-padded Gaussian blur
// on 8 x 1024 x 1024 f32. Fused single kernel, LDS staged via gfx1250 async
// global->LDS loads (ASYNCcnt path), separable Gaussian, float4 LDS/global I/O.

#define IMG_H 1024
#define IMG_W 1024
#define TILE 64
#define IN_DIM 72    // TILE + 2*4 halo (median radius 2 + gauss radius 2)
#define MED_DIM 68   // TILE + 2*2 halo (gauss radius 2)
#define NTHREADS 256

__device__ __forceinline__ int reflect_idx(int i, int n) {
    // jnp.pad mode='reflect' (no edge repeat): -1 -> 1, -2 -> 2 ; n -> n-2 ...
    if (i < 0) i = -i;
    if (i >= n) i = 2 * n - 2 - i;
    return i;
}

// compare-exchange: a <- min, b <- max
#define S2(a, b)                                   \
    {                                              \
        float t_ = fminf(p[a], p[b]);              \
        p[b] = fmaxf(p[a], p[b]);                  \
        p[a] = t_;                                 \
    }

// Devillard opt_med25: 99-exchange selection network, median lands in p[12].
__device__ __forceinline__ float median25(float p[25]) {
    S2(0,1)  S2(3,4)  S2(2,4)  S2(2,3)  S2(6,7)  S2(5,7)  S2(5,6)  S2(9,10)
    S2(8,10) S2(8,9)  S2(12,13) S2(11,13) S2(11,12) S2(15,16) S2(14,16)
    S2(14,15) S2(18,19) S2(17,19) S2(17,18) S2(21,22) S2(20,22) S2(20,21)
    S2(23,24) S2(2,5)  S2(3,6)  S2(0,6)  S2(0,3)  S2(4,7)  S2(1,7)  S2(1,4)
    S2(11,14) S2(8,14) S2(8,11) S2(12,15) S2(9,15) S2(9,12) S2(13,16)
    S2(10,16) S2(10,13) S2(20,23) S2(17,23) S2(17,20) S2(21,24) S2(18,24)
    S2(18,21) S2(19,22) S2(8,17) S2(9,18) S2(0,18) S2(0,9)  S2(10,19)
    S2(1,19) S2(1,10) S2(11,20) S2(2,20) S2(2,11) S2(12,21) S2(3,21)
    S2(3,12) S2(13,22) S2(4,22) S2(4,13) S2(14,23) S2(5,23) S2(5,14)
    S2(15,24) S2(6,24) S2(6,15) S2(7,16) S2(7,19) S2(13,21) S2(15,23)
    S2(7,13) S2(7,15) S2(1,9)  S2(3,11) S2(5,17) S2(11,17) S2(9,17)
    S2(4,10) S2(6,12) S2(7,14) S2(4,6)  S2(4,7)  S2(12,14) S2(10,14)
    S2(6,7)  S2(10,12) S2(6,10) S2(6,17) S2(12,17) S2(7,17) S2(7,10)
    S2(12,18) S2(7,12) S2(10,18) S2(12,20) S2(10,20) S2(10,12)
    return p[12];
}

__global__ __launch_bounds__(NTHREADS)
void smoothing_fused_kernel(const float* __restrict__ img,
                            const float* __restrict__ sigma_p,
                            float* __restrict__ out)
{
    __shared__ float ldsIn[IN_DIM * IN_DIM];    // input tile; reused as H-blur buffer
    __shared__ float ldsMed[MED_DIM * MED_DIM]; // median tile (0 outside image)

    const int tid = threadIdx.x;
    const int X0 = blockIdx.x * TILE;
    const int Y0 = blockIdx.y * TILE;
    const int b  = blockIdx.z;
    const float* src = img + (size_t)b * (IMG_H * IMG_W);

    // ---------- Stage A: async global -> LDS load of reflected 72x72 tile ----------
    const uint32_t ldsInBase = (uint32_t)(uintptr_t)(&ldsIn[0]); // low 32 bits = LDS offset
    for (int idx = tid; idx < IN_DIM * IN_DIM; idx += NTHREADS) {
        const int yy = idx / IN_DIM;
        const int xx = idx - yy * IN_DIM;
        const int gy = reflect_idx(Y0 - 4 + yy, IMG_H);
        const int gx = reflect_idx(X0 - 4 + xx, IMG_W);
        const float* gptr = src + (size_t)gy * IMG_W + gx;
        const uint32_t loff = ldsInBase + (uint32_t)idx * 4u;
        // CDNA5 async DMA element copy: copies 4 bytes from the global address in gptr to the LDS address in loff, tracked by ASYNCcnt
        asm volatile("global_load_async_to_lds_b32 %0, %1, off"
                     :: "v"(loff), "v"(gptr)
                     : "memory");
    }
    asm volatile("s_wait_asynccnt 0" ::: "memory");
    __syncthreads();

    // ---------- Gaussian 1-D weights from sigma (separable, normalized) ----------
    const float sg = sigma_p[0];
    const float inv2s2 = 1.0f / (2.0f * sg * sg);
    const float w2 = __expf(-4.0f * inv2s2);
    const float w1 = __expf(-1.0f * inv2s2);
    const float w0 = 1.0f;
    const float rs = 1.0f / (w0 + 2.0f * w1 + 2.0f * w2);
    const float k0 = w2 * rs;  // offset -2 / +2
    const float k1 = w1 * rs;  // offset -1 / +1
    const float k2 = w0 * rs;  // center

    // ---------- Stage B: 5x5 median over 68x68 region (zero outside image) ----------
    for (int idx = tid; idx < MED_DIM * MED_DIM; idx += NTHREADS) {
        const int my = idx / MED_DIM;
        const int mx = idx - my * MED_DIM;
        const int gy = Y0 - 2 + my;
        const int gx = X0 - 2 + mx;
        float m = 0.0f;
        if (gy >= 0 && gy < IMG_H && gx >= 0 && gx < IMG_W) {
            float p[25];
#pragma unroll
            for (int dy = 0; dy < 5; ++dy) {
#pragma unroll
                for (int dx = 0; dx < 5; ++dx) {
                    p[dy * 5 + dx] = ldsIn[(my + dy) * IN_DIM + (mx + dx)];
                }
            }
            m = median25(p);
        }
        ldsMed[idx] = m;
    }
    __syncthreads();

    // ---------- Stage C: horizontal 5-tap Gaussian, float4 (68 rows x 16 groups) ----
    float* hb = ldsIn; // reuse input tile as the H-blurred buffer (68 x 64)
    for (int idx = tid; idx < MED_DIM * (TILE / 4); idx += NTHREADS) {
        const int r  = idx >> 4;        // row 0..67
        const int cg = idx & 15;        // group of 4 output cols
        // need med[r][4cg .. 4cg+7]: two aligned float4 loads (row stride 68 = 17*4)
        const float4* mrow = reinterpret_cast<const float4*>(&ldsMed[r * MED_DIM + 4 * cg]);
        const float4 a = mrow[0];
        const float4 bq = mrow[1];
        float4 v;
        v.x = k0 * a.x + k1 * a.y + k2 * a.z + k1 * a.w + k0 * bq.x;
        v.y = k0 * a.y + k1 * a.z + k2 * a.w + k1 * bq.x + k0 * bq.y;
        v.z = k0 * a.z + k1 * a.w + k2 * bq.x + k1 * bq.y + k0 * bq.z;
        v.w = k0 * a.w + k1 * bq.x + k2 * bq.y + k1 * bq.z + k0 * bq.w;
        reinterpret_cast<float4*>(hb)[idx] = v;   // hb[r][4cg..4cg+3], aligned
    }
    __syncthreads();

    // ---------- Stage D: vertical 5-tap Gaussian, float4 output stores ----------
    float* dstb = out + (size_t)b * (IMG_H * IMG_W);
    for (int idx = tid; idx < TILE * (TILE / 4); idx += NTHREADS) {  // 1024 = 4*256
        const int y  = idx >> 4;        // 0..63
        const int xg = idx & 15;        // group of 4 cols
        const float4 r0 = reinterpret_cast<const float4*>(&hb[(y + 0) * TILE])[xg];
        const float4 r1 = reinterpret_cast<const float4*>(&hb[(y + 1) * TILE])[xg];
        const float4 r2 = reinterpret_cast<const float4*>(&hb[(y + 2) * TILE])[xg];
        const float4 r3 = reinterpret_cast<const float4*>(&hb[(y + 3) * TILE])[xg];
        const float4 r4 = reinterpret_cast<const float4*>(&hb[(y + 4) * TILE])[xg];
        float4 v;
        v.x = k0 * r0.x + k1 * r1.x + k2 * r2.x + k1 * r3.x + k0 * r4.x;
        v.y = k0 * r0.y + k1 * r1.y + k2 * r2.y + k1 * r3.y + k0 * r4.y;
        v.z = k0 * r0.z + k1 * r1.z + k2 * r2.z + k1 * r3.z + k0 * r4.z;
        v.w = k0 * r0.w + k1 * r1.w + k2 * r2.w + k1 * r3.w + k0 * r4.w;
        // (Y0+y, X0+4*xg): X0 multiple of 64 -> 16B aligned global b128 store
        float4* drow = reinterpret_cast<float4*>(dstb + (size_t)(Y0 + y) * IMG_W + X0);
        drow[xg] = v;
    }
}

extern "C" void kernel_launch(void* const* d_in, const int* in_sizes, int n_in,
                              void* d_out, int out_size, void* d_ws, size_t ws_size,
                              hipStream_t stream) {
    (void)n_in; (void)d_ws; (void)ws_size; (void)out_size;
    const float* img   = (const float*)d_in[0];
    const float* sigma = (const float*)d_in[1];
    float* out = (float*)d_out;

    const int B = in_sizes[0] / (IMG_H * IMG_W); // 8
    dim3 grid(IMG_W / TILE, IMG_H / TILE, B);
    dim3 block(NTHREADS, 1, 1);
    smoothing_fused_kernel<<<grid, block, 0, stream>>>(img, sigma, out);
}